// SWD7_28449863369551
// MI455X (gfx1250) — compile-verified
//
#include <hip/hip_runtime.h>
#include <hip/hip_bf16.h>
#include <cmath>

// Shapes from the reference: B=4, H=16, L=8192, D=64 (fp32). q,k are UNUSED.
#define BH_N 64
#define L_N  8192
#define D_N  64
#define CHP  32                    // chunks per (b,h) in the fused pass
#define ROWSP (L_N / CHP)          // 256 rows per block
#define ITERS (ROWSP / 16)         // 16 pipeline iterations (16 rows each)
#define NV  ((size_t)BH_N * L_N * D_N)   // 33,554,432 elements per output copy

typedef float v4f __attribute__((ext_vector_type(4)));
typedef int   v4i __attribute__((ext_vector_type(4)));

// ---- CDNA5 async global->LDS DMA (ASYNCcnt-tracked), inline asm: portable
// ---- across ROCm 7.2 and the amdgpu-toolchain lane.
__device__ __forceinline__ void async_ld_b128(unsigned lds_addr, const void* gaddr) {
  asm volatile("global_load_async_to_lds_b128 %0, %1, off"
               :
               : "v"(lds_addr), "v"(gaddr)
               : "memory");
}
#define WAIT_ASYNC(n) asm volatile("s_wait_asynccnt " #n ::: "memory")

// ---- Phase 1 (fused): stream v through LDS via 4-deep async pipeline;
// ----   emit both output copies (NT) and per-(bh,chunk) partial max/argmax.
__global__ void __launch_bounds__(256)
p1_fused(const float* __restrict__ v, float* __restrict__ out,
         float* __restrict__ pmax, int* __restrict__ pidx) {
  const int bh  = blockIdx.x / CHP;
  const int ck  = blockIdx.x % CHP;
  const int tid = threadIdx.x;
  const int col = (tid & 15) << 2;   // 4 columns per thread
  const int rl  = tid >> 4;          // 16 row-lanes
  const size_t vbase = (size_t)bh * L_N * D_N;
  const int l0 = ck * ROWSP;

  __shared__ v4f tile[4][256];
  unsigned lds_a[4];
  lds_a[0] = (unsigned)(size_t)&tile[0][tid];  // generic LDS addr: low 32 bits = DS offset
  lds_a[1] = (unsigned)(size_t)&tile[1][tid];
  lds_a[2] = (unsigned)(size_t)&tile[2][tid];
  lds_a[3] = (unsigned)(size_t)&tile[3][tid];

  auto gsrc = [&](int it) -> const void* {
    return (const void*)(v + vbase + (size_t)(l0 + it * 16 + rl) * D_N + col);
  };

  v4f m  = { -INFINITY, -INFINITY, -INFINITY, -INFINITY };
  v4i mi = { 0, 0, 0, 0 };

  // ascending l per thread + strict '>' => first-occurrence within thread subset
  auto process = [&](int it) {
    v4f x = tile[it & 3][tid];                 // own slot; no cross-thread share
    const int lc = l0 + it * 16 + rl;
    if (x.x > m.x) { m.x = x.x; mi.x = lc; }
    if (x.y > m.y) { m.y = x.y; mi.y = lc; }
    if (x.z > m.z) { m.z = x.z; mi.z = lc; }
    if (x.w > m.w) { m.w = x.w; mi.w = lc; }
    const size_t off = vbase + (size_t)lc * D_N + col;
    __builtin_nontemporal_store(x, (v4f*)(out + off));        // copy 1 (NT: spare L2)
    __builtin_nontemporal_store(x, (v4f*)(out + NV + off));   // copy 2
  };

  // 4-deep software pipeline on the async DMA engine
  async_ld_b128(lds_a[0], gsrc(0));
  async_ld_b128(lds_a[1], gsrc(1));
  async_ld_b128(lds_a[2], gsrc(2));
#pragma unroll
  for (int it = 0; it < ITERS - 3; ++it) {
    async_ld_b128(lds_a[(it + 3) & 3], gsrc(it + 3));
    WAIT_ASYNC(3);                             // in-order completion: iter 'it' landed
    process(it);
  }
  WAIT_ASYNC(2); process(ITERS - 3);
  WAIT_ASYNC(1); process(ITERS - 2);
  WAIT_ASYNC(0); process(ITERS - 1);

  // ---- block-level reduction of (m, mi) over the 16 row-lanes ----
  __shared__ v4f sm[256];
  __shared__ v4i si[256];
  sm[tid] = m; si[tid] = mi;
  __syncthreads();
  for (int s = 128; s >= 16; s >>= 1) {        // stride multiples of 16 keep columns aligned
    if (tid < s) {
      v4f om = sm[tid + s]; v4i oi = si[tid + s];
      v4f cm = sm[tid];     v4i ci = si[tid];
      if (om.x > cm.x || (om.x == cm.x && oi.x < ci.x)) { cm.x = om.x; ci.x = oi.x; }
      if (om.y > cm.y || (om.y == cm.y && oi.y < ci.y)) { cm.y = om.y; ci.y = oi.y; }
      if (om.z > cm.z || (om.z == cm.z && oi.z < ci.z)) { cm.z = om.z; ci.z = oi.z; }
      if (om.w > cm.w || (om.w == cm.w && oi.w < ci.w)) { cm.w = om.w; ci.w = oi.w; }
      sm[tid] = cm; si[tid] = ci;
    }
    __syncthreads();
  }
  if (tid < 16) {
    const size_t o = ((size_t)bh * CHP + ck) * D_N + (tid << 2);
    *(v4f*)(pmax + o) = sm[tid];
    *(v4i*)(pidx + o) = si[tid];
  }
}

// ---- Phase 2: combine chunk partials (ascending-l order keeps first occurrence)
__global__ void __launch_bounds__(64)
p2_final(const float* __restrict__ pmax, const int* __restrict__ pidx,
         float* __restrict__ fmax, int* __restrict__ fidx) {
  const int bh = blockIdx.x;
  const int d  = threadIdx.x;
  float m = -INFINITY; int mi = 0;
#pragma unroll 4
  for (int c = 0; c < CHP; ++c) {
    const size_t o = ((size_t)bh * CHP + c) * D_N + d;
    const float pm = pmax[o];
    if (pm > m) { m = pm; mi = pidx[o]; }
  }
  fmax[bh * D_N + d] = m;
  fidx[bh * D_N + d] = mi;
}

// ---- Phase 3: tiny fixup over the already-copied output -------------------
//   row 0 := column maxes, then row argmax[d] := v_cls[d] (overrides row 0;
//   when idx==0 both values are numerically identical anyway).
__global__ void __launch_bounds__(64)
p3_fixup(const float* __restrict__ v, const float* __restrict__ fmax,
         const int* __restrict__ fidx, float* __restrict__ out) {
  const int bh = blockIdx.x;
  const int d  = threadIdx.x;
  const size_t vbase = (size_t)bh * L_N * D_N;
  const float m   = fmax[bh * D_N + d];
  const int   idx = fidx[bh * D_N + d];
  const float vc  = v[vbase + d];              // v_cls = v[b,h,0,d]

  out[vbase + d]      = m;                     // row 0, copy 1
  out[NV + vbase + d] = m;                     // row 0, copy 2
  const size_t so = vbase + (size_t)idx * D_N + d;
  out[so]      = vc;                           // scatter, copy 1
  out[NV + so] = vc;                           // scatter, copy 2
}

extern "C" void kernel_launch(void* const* d_in, const int* in_sizes, int n_in,
                              void* d_out, int out_size, void* d_ws, size_t ws_size,
                              hipStream_t stream) {
  (void)in_sizes; (void)n_in; (void)out_size; (void)ws_size;
  const float* v = (const float*)d_in[2];      // inputs: q, k, v (q,k unused)
  float* out = (float*)d_out;

  float* pmax = (float*)d_ws;                                  // BH*CHP*D floats (512 KB)
  int*   pidx = (int*)(pmax + (size_t)BH_N * CHP * D_N);       // BH*CHP*D ints  (512 KB)
  float* fmax = (float*)(pidx + (size_t)BH_N * CHP * D_N);     // BH*D floats
  int*   fidx = (int*)(fmax + (size_t)BH_N * D_N);             // BH*D ints

  p1_fused<<<BH_N * CHP, 256, 0, stream>>>(v, out, pmax, pidx);
  p2_final<<<BH_N,        64, 0, stream>>>(pmax, pidx, fmax, fidx);
  p3_fixup<<<BH_N,        64, 0, stream>>>(v, fmax, fidx, out);
}